// InpEmbed_9088150798985
// MI455X (gfx1250) — compile-verified
//
#include <hip/hip_runtime.h>
#include <math.h>

// ---------------------------------------------------------------------------
// InpEmbed: out[t, :] = table[x[t], :] + pos[:]
//   x:     [64*1024]  int64 token ids (block-uniform per token -> s_load)
//   table: [32000, 512] fp32 (64 MiB -> L2-resident on MI455X's 192 MB L2)
//   out:   [64*1024, 512] fp32 (128 MiB, write-once -> non-temporal stores)
//
// Memory-bound gather: ~256 MiB of traffic, roofline ~11 us @ 23.3 TB/s.
// The positional vector is a 512-float constant: computing it per-wave with
// libm cost ~700 VALU instrs/wave (~9 us machine-wide) -- so it is computed
// ONCE into d_ws by a 1-block kernel, and the gather kernel just loads it.
// ---------------------------------------------------------------------------

#define DEMBED            512
#define THREADS           128   // 4 wave32s; one float4 per lane covers a row
#define TOKENS_PER_BLOCK  8

typedef __attribute__((ext_vector_type(4))) float f4;

// ---- kernel 1: pos[512] -> d_ws (runs as a single block, negligible) ------
__global__ __launch_bounds__(THREADS) void InpEmbed_pos_init_kernel(
    float* __restrict__ pos_out)
{
  const int i = (int)threadIdx.x * 4;
#pragma unroll
  for (int j = 0; j < 4; ++j) {
    const int   k  = i + j;
    const int   e  = k & ~1;                                  // floor(k/2)*2
    const float ep = powf(10000.0f, -(float)e / (float)DEMBED);
    const float p  = (float)k * ep;
    pos_out[k] = (k & 1) ? cosf(p) : sinf(p);
  }
}

// ---- kernel 2: streaming gather + add -------------------------------------
__global__ __launch_bounds__(THREADS) void InpEmbed_gather_pos_kernel(
    const long long* __restrict__ x,     // [ntok] int64 indices
    const float*     __restrict__ table, // [VOCAB, DEMBED] fp32
    const float*     __restrict__ pos,   // [DEMBED] fp32 (precomputed)
    float*           __restrict__ out,   // [ntok, DEMBED] fp32
    int ntok)
{
  // one float4 of the positional vector per lane (L2-hit, 2 KiB total)
  const f4 pos4 = ((const f4*)pos)[threadIdx.x];

  const long long base_tok = (long long)blockIdx.x * TOKENS_PER_BLOCK;

  if (base_tok + TOKENS_PER_BLOCK <= (long long)ntok) {
    // batch the block-uniform index loads (scalar s_load, all in flight) ...
    long long idxs[TOKENS_PER_BLOCK];
#pragma unroll
    for (int u = 0; u < TOKENS_PER_BLOCK; ++u)
      idxs[u] = x[base_tok + u];

    // ... then stream rows: b128 load -> packed add -> non-temporal b128 store
#pragma unroll
    for (int u = 0; u < TOKENS_PER_BLOCK; ++u) {
      const long long t   = base_tok + u;
      const f4*       row = (const f4*)(table + idxs[u] * (long long)DEMBED);
      f4 v = row[threadIdx.x];             // RT load: keep table in L2
      v    = v + pos4;
      __builtin_nontemporal_store(v, (f4*)(out + t * (long long)DEMBED) + threadIdx.x);
    }
  } else {
    // tail (not hit for 64*1024 tokens, kept for generality)
    for (long long t = base_tok; t < (long long)ntok; ++t) {
      const long long idx = x[t];
      const f4*       row = (const f4*)(table + idx * (long long)DEMBED);
      f4 v = row[threadIdx.x];
      v    = v + pos4;
      __builtin_nontemporal_store(v, (f4*)(out + t * (long long)DEMBED) + threadIdx.x);
    }
  }
}

extern "C" void kernel_launch(void* const* d_in, const int* in_sizes, int n_in,
                              void* d_out, int out_size, void* d_ws, size_t ws_size,
                              hipStream_t stream) {
  const long long* x     = (const long long*)d_in[0]; // int64 token ids
  const float*     table = (const float*)d_in[1];     // fp32 embedding table
  float*           out   = (float*)d_out;             // fp32 output
  float*           pos   = (float*)d_ws;              // 2 KiB scratch

  const int ntok   = in_sizes[0];                         // 64*1024 = 65536
  const int blocks = (ntok + TOKENS_PER_BLOCK - 1) / TOKENS_PER_BLOCK; // 8192

  InpEmbed_pos_init_kernel<<<1, THREADS, 0, stream>>>(pos);
  InpEmbed_gather_pos_kernel<<<blocks, THREADS, 0, stream>>>(x, table, pos, out, ntok);
}